// GCN_89696097009669
// MI455X (gfx1250) — compile-verified
//
#include <hip/hip_runtime.h>
#include <math.h>

#define N_NODES 100000
#define N_EDGES 1600000
#define NFEAT 256
#define HIDDEN 128
#define NCLASS 40

typedef __attribute__((ext_vector_type(2))) float v2f;
typedef __attribute__((ext_vector_type(8))) float v8f;

// ---------------------------------------------------------------------------
// GEMM1: H1[N_NODES,128] = X[N_NODES,256] @ W1[256,128], fp32 WMMA 16x16x4.
// Block: 256 threads = 8 waves; wave w computes the 16x16 tile at col 16*w.
// Grid.x: N_NODES/16 = 6250 row tiles.
// ---------------------------------------------------------------------------
__global__ __launch_bounds__(256) void gcn_gemm1(const float* __restrict__ X,
                                                 const float* __restrict__ W1,
                                                 float* __restrict__ H1) {
    const int lane  = threadIdx.x & 31;
    const int wave  = threadIdx.x >> 5;          // 0..7 -> column tile
    const int m     = lane & 15;                 // row within tile
    const int khalf = lane >> 4;                 // 0 or 1: K sub-pair
    const int n     = wave * 16 + m;             // output column
    const size_t row0 = (size_t)blockIdx.x * 16;

    const float* xrow = X + (row0 + m) * NFEAT;
    v8f acc = {};
#pragma unroll 4
    for (int k0 = 0; k0 < NFEAT; k0 += 4) {
        const int ka = k0 + 2 * khalf;
        v2f a, b;
        a.x = xrow[ka];
        a.y = xrow[ka + 1];
        b.x = W1[(size_t)ka * HIDDEN + n];
        b.y = W1[(size_t)(ka + 1) * HIDDEN + n];
        acc = __builtin_amdgcn_wmma_f32_16x16x4_f32(false, a, false, b,
                                                    (short)0, acc, false, false);
    }
    // C/D layout: VGPR v holds M=v (lanes 0-15) / M=v+8 (lanes 16-31), N=lane%16
    float* orow = H1 + (row0 + 8 * khalf) * HIDDEN + n;
#pragma unroll
    for (int v = 0; v < 8; ++v)
        orow[(size_t)v * HIDDEN] = acc[v];
}

// ---------------------------------------------------------------------------
// GEMM2: Z[N_NODES,40] = H[N_NODES,128] @ W2[128,40], fp32 WMMA, 3 col tiles
// (48 cols, stores masked to 40). Block: 96 threads = 3 waves.
// ---------------------------------------------------------------------------
__global__ __launch_bounds__(96) void gcn_gemm2(const float* __restrict__ H,
                                                const float* __restrict__ W2,
                                                float* __restrict__ Z) {
    const int lane  = threadIdx.x & 31;
    const int wave  = threadIdx.x >> 5;          // 0..2 -> column tile
    const int m     = lane & 15;
    const int khalf = lane >> 4;
    const int n     = wave * 16 + m;             // 0..47
    const size_t row0 = (size_t)blockIdx.x * 16;

    const float* hrow = H + (row0 + m) * HIDDEN;
    const bool nok = (n < NCLASS);
    v8f acc = {};
#pragma unroll 4
    for (int k0 = 0; k0 < HIDDEN; k0 += 4) {
        const int ka = k0 + 2 * khalf;
        v2f a, b;
        a.x = hrow[ka];
        a.y = hrow[ka + 1];
        b.x = nok ? W2[(size_t)ka * NCLASS + n] : 0.0f;
        b.y = nok ? W2[(size_t)(ka + 1) * NCLASS + n] : 0.0f;
        acc = __builtin_amdgcn_wmma_f32_16x16x4_f32(false, a, false, b,
                                                    (short)0, acc, false, false);
    }
    if (nok) {
        float* orow = Z + (row0 + 8 * khalf) * NCLASS + n;
#pragma unroll
        for (int v = 0; v < 8; ++v)
            orow[(size_t)v * NCLASS] = acc[v];
    }
}

// ---------------------------------------------------------------------------
// SpMM layer 1: acc[dst] += w * h1[src], 128 feats. Wave per edge; lane owns
// 4 consecutive feats (float4 gather, 4x native f32 atomic scatter).
// unsafeAtomicAdd -> GLOBAL_ATOMIC_ADD_F32 (no CAS loop).
// ---------------------------------------------------------------------------
__global__ __launch_bounds__(256) void gcn_spmm1(const int* __restrict__ esrc,
                                                 const int* __restrict__ edst,
                                                 const float* __restrict__ ew,
                                                 const float* __restrict__ h1,
                                                 float* __restrict__ acc) {
    const long long tid = (long long)blockIdx.x * blockDim.x + threadIdx.x;
    const int e = (int)(tid >> 5);
    if (e >= N_EDGES) return;
    const int q = (int)(tid & 31);               // feat quad
    const int s = esrc[e];
    const int d = edst[e];
    const float w = ew[e];
    const float4 v = *(const float4*)(h1 + (size_t)s * HIDDEN + q * 4);
    float* o = acc + (size_t)d * HIDDEN + q * 4;
    unsafeAtomicAdd(o + 0, w * v.x);
    unsafeAtomicAdd(o + 1, w * v.y);
    unsafeAtomicAdd(o + 2, w * v.z);
    unsafeAtomicAdd(o + 3, w * v.w);
}

// ---------------------------------------------------------------------------
// SpMM layer 2: out[dst] += w * z[src], 40 feats. Thread per (edge,class).
// ---------------------------------------------------------------------------
__global__ __launch_bounds__(256) void gcn_spmm2(const int* __restrict__ esrc,
                                                 const int* __restrict__ edst,
                                                 const float* __restrict__ ew,
                                                 const float* __restrict__ z,
                                                 float* __restrict__ out) {
    const long long tid = (long long)blockIdx.x * blockDim.x + threadIdx.x;
    if (tid >= (long long)N_EDGES * NCLASS) return;
    const int e = (int)(tid / NCLASS);
    const int c = (int)(tid - (long long)e * NCLASS);
    const int s = esrc[e];
    const int d = edst[e];
    unsafeAtomicAdd(out + (size_t)d * NCLASS + c,
                    ew[e] * z[(size_t)s * NCLASS + c]);
}

// ---------------------------------------------------------------------------
// bias + relu + dropout (exact JAX threefry2x32-20 with key (0, 42)).
// ---------------------------------------------------------------------------
__device__ __forceinline__ unsigned rotl32(unsigned x, int d) {
    return (x << d) | (x >> (32 - d));
}

__device__ __forceinline__ void threefry2x32(unsigned k0, unsigned k1,
                                             unsigned x0, unsigned x1,
                                             unsigned* o0, unsigned* o1) {
    const unsigned ks2 = k0 ^ k1 ^ 0x1BD11BDAu;
    x0 += k0; x1 += k1;
#define TF_GROUP(RA, RB, RC, RD, KA, KB, INC)                                  \
    x0 += x1; x1 = rotl32(x1, RA); x1 ^= x0;                                   \
    x0 += x1; x1 = rotl32(x1, RB); x1 ^= x0;                                   \
    x0 += x1; x1 = rotl32(x1, RC); x1 ^= x0;                                   \
    x0 += x1; x1 = rotl32(x1, RD); x1 ^= x0;                                   \
    x0 += (KA); x1 += (KB) + (INC);
    TF_GROUP(13, 15, 26,  6, k1,  ks2, 1u)
    TF_GROUP(17, 29, 16, 24, ks2, k0,  2u)
    TF_GROUP(13, 15, 26,  6, k0,  k1,  3u)
    TF_GROUP(17, 29, 16, 24, k1,  ks2, 4u)
    TF_GROUP(13, 15, 26,  6, ks2, k0,  5u)
#undef TF_GROUP
    *o0 = x0; *o1 = x1;
}

__global__ __launch_bounds__(256) void gcn_relu_dropout(float* __restrict__ h,
                                                        const float* __restrict__ b1) {
    const unsigned i = blockIdx.x * 256u + threadIdx.x;
    const unsigned TOT  = (unsigned)N_NODES * HIDDEN;   // 12,800,000
    const unsigned HALF = TOT / 2u;
    if (i >= TOT) return;
    unsigned x0, x1;
    const bool first = (i < HALF);
    if (first) { x0 = i;        x1 = i + HALF; }
    else       { x0 = i - HALF; x1 = i;        }
    unsigned o0, o1;
    threefry2x32(0u, 42u, x0, x1, &o0, &o1);
    const unsigned bits = first ? o0 : o1;
    const float u = __uint_as_float((bits >> 9) | 0x3f800000u) - 1.0f;
    float v = h[i] + b1[i & (HIDDEN - 1)];
    v = v > 0.0f ? v : 0.0f;                     // relu
    h[i] = (u < 0.5f) ? v * 2.0f : 0.0f;         // inverted dropout, keep=0.5
}

// ---------------------------------------------------------------------------
// bias + row log_softmax over 40 classes, wave per row (wave32 shfl reduce).
// ---------------------------------------------------------------------------
__global__ __launch_bounds__(256) void gcn_logsoftmax(float* __restrict__ out,
                                                      const float* __restrict__ b2) {
    const int lane = threadIdx.x & 31;
    const int row  = blockIdx.x * 8 + (threadIdx.x >> 5);
    if (row >= N_NODES) return;
    float* r = out + (size_t)row * NCLASS;

    const int c2 = lane + 32;
    const bool ok2 = (c2 < NCLASS);
    float v1 = r[lane] + b2[lane];
    float v2 = ok2 ? (r[c2] + b2[c2]) : -INFINITY;

    float m = fmaxf(v1, v2);
#pragma unroll
    for (int off = 16; off; off >>= 1)
        m = fmaxf(m, __shfl_xor(m, off, 32));

    float s = __expf(v1 - m) + (ok2 ? __expf(v2 - m) : 0.0f);
#pragma unroll
    for (int off = 16; off; off >>= 1)
        s += __shfl_xor(s, off, 32);

    const float l = m + __logf(s);
    r[lane] = v1 - l;
    if (ok2) r[c2] = v2 - l;
}

// ---------------------------------------------------------------------------
extern "C" void kernel_launch(void* const* d_in, const int* in_sizes, int n_in,
                              void* d_out, int out_size, void* d_ws, size_t ws_size,
                              hipStream_t stream) {
    const float* x    = (const float*)d_in[0];
    const int*   esrc = (const int*)  d_in[1];
    const int*   edst = (const int*)  d_in[2];
    const float* ew   = (const float*)d_in[3];
    const float* W1   = (const float*)d_in[4];
    const float* b1   = (const float*)d_in[5];
    const float* W2   = (const float*)d_in[6];
    const float* b2   = (const float*)d_in[7];
    float* out = (float*)d_out;

    float* h1   = (float*)d_ws;                          // [N_NODES,128]
    float* hacc = h1   + (size_t)N_NODES * HIDDEN;       // [N_NODES,128]
    float* z    = hacc + (size_t)N_NODES * HIDDEN;       // [N_NODES,40]

    hipMemsetAsync(hacc, 0, (size_t)N_NODES * HIDDEN * sizeof(float), stream);

    // Layer 1 dense: h1 = x @ W1
    gcn_gemm1<<<N_NODES / 16, 256, 0, stream>>>(x, W1, h1);

    // Layer 1 sparse: hacc = A @ h1
    {
        long long thr = (long long)N_EDGES * 32;
        gcn_spmm1<<<(unsigned)((thr + 255) / 256), 256, 0, stream>>>(
            esrc, edst, ew, h1, hacc);
    }

    // bias + relu + dropout in place
    gcn_relu_dropout<<<(N_NODES * HIDDEN + 255) / 256, 256, 0, stream>>>(hacc, b1);

    // Layer 2 dense: z = hacc @ W2
    gcn_gemm2<<<N_NODES / 16, 96, 0, stream>>>(hacc, W2, z);

    // Layer 2 sparse into d_out
    hipMemsetAsync(out, 0, (size_t)N_NODES * NCLASS * sizeof(float), stream);
    {
        long long thr = (long long)N_EDGES * NCLASS;
        gcn_spmm2<<<(unsigned)((thr + 255) / 256), 256, 0, stream>>>(
            esrc, edst, ew, z, out);
    }

    // bias + log_softmax in place
    gcn_logsoftmax<<<(N_NODES + 7) / 8, 256, 0, stream>>>(out, b2);
}